// SparseRouter_66571993088219
// MI455X (gfx1250) — compile-verified
//
#include <hip/hip_runtime.h>
#include <math.h>

// ---------------------------------------------------------------------------
// SparseRouter for MI455X (gfx1250, wave32).
// B=131072 rows, D=1024, E=64 experts, K=2.
// Memory-bound (~570MB traffic => ~24us @ 23.3TB/s). GEMM in full f32 via
// V_WMMA_F32_16X16X4_F32 (precision matches reference; bandwidth is the
// ceiling, not matrix math). Staging uses GLOBAL_LOAD_ASYNC_TO_LDS_B128 with
// double buffering so DMA overlaps the WMMA stream (ASYNCcnt pipeline).
// ---------------------------------------------------------------------------

typedef __attribute__((ext_vector_type(2))) float v2f;
typedef __attribute__((ext_vector_type(8))) float v8f;
typedef __attribute__((ext_vector_type(4))) int   v4i;

#define NB       131072
#define ND       1024
#define NE       64
#define ROWS_PB  128          // rows per block
#define KC       32           // K-chunk staged in LDS (double buffered)
#define NCHUNK   (ND / KC)    // 32 chunks
#define XST      (KC + 4)     // padded strides: bank-conflict-free frag reads
#define WST      (KC + 4)
#define EST      65           // epilogue stride (odd => conflict-free col sums)
#define BETA_F   1.0e6f
#define NEG_INF  (-3.402823466e38f)
#define OPS_PT   8            // async b128 ops issued per thread per chunk

#if defined(__HIP_DEVICE_COMPILE__) && \
    __has_builtin(__builtin_amdgcn_global_load_async_to_lds_b128) && \
    __has_builtin(__builtin_amdgcn_s_wait_asynccnt)
#define USE_ASYNC_LDS 1
#else
#define USE_ASYNC_LDS 0
#endif

// GEMM staging (2 buffers) and epilogue arrays are disjoint in time -> union.
union SMem {
    struct {                              // 2 x 36,864 B = 73,728 B
        float X[2][ROWS_PB * XST];        // 128 x (32+4)
        float Wg[2][NE * WST];            // 64 x (32+4)
        float Ws[2][NE * WST];
    } g;
    struct {                              // 66,560 B
        float H[ROWS_PB * EST];           // h
        float P[ROWS_PB * EST];           // scale -> P -> gates
    } e;
};

__device__ __forceinline__ void copy16_to_lds(const float* gsrc, float* ldst) {
#if USE_ASYNC_LDS
    // GLOBAL_LOAD_ASYNC_TO_LDS_B128: memory -> LDS direct, tracked by ASYNCcnt
    __builtin_amdgcn_global_load_async_to_lds_b128((v4i*)gsrc, (v4i*)ldst, 0, 0);
#else
    *(float4*)ldst = *(const float4*)gsrc;
#endif
}

__device__ __forceinline__ void stage_chunk(const float* __restrict__ x,
                                            const float* __restrict__ Wg,
                                            const float* __restrict__ Ws,
                                            long row0, int kc,
                                            float* sX, float* sWg, float* sWs,
                                            int tid) {
    // x tile: 128 rows x KC cols = 1024 float4 => 4 per thread, coalesced
    #pragma unroll
    for (int i = tid; i < ROWS_PB * (KC / 4); i += 256) {
        const int r  = i / (KC / 4);
        const int c4 = (i % (KC / 4)) * 4;
        copy16_to_lds(&x[(row0 + r) * (long)ND + kc + c4], &sX[r * XST + c4]);
    }
    // W_g / W_s chunks: 64 x KC each = 512 float4 => 2+2 per thread
    #pragma unroll
    for (int i = tid; i < NE * (KC / 4); i += 256) {
        const int r  = i / (KC / 4);
        const int c4 = (i % (KC / 4)) * 4;
        copy16_to_lds(&Wg[r * ND + kc + c4], &sWg[r * WST + c4]);
        copy16_to_lds(&Ws[r * ND + kc + c4], &sWs[r * WST + c4]);
    }
}

__global__ void router_zero_ws(float* ws) {
    if (threadIdx.x < 2 * NE) ws[threadIdx.x] = 0.0f;
}

__launch_bounds__(256, 1)
__global__ void router_main(const float* __restrict__ x,
                            const float* __restrict__ Wg,
                            const float* __restrict__ Ws,
                            const float* __restrict__ eps,
                            float* __restrict__ gates,
                            float* __restrict__ wsum) {   // [0..63]=h colsum, [64..127]=P colsum
    __shared__ SMem sm;

    const int  tid   = threadIdx.x;
    const int  lane  = tid & 31;
    const int  wave  = tid >> 5;       // 8 waves, each owns 16 rows
    const int  m16   = lane & 15;
    const int  khalf = lane >> 4;      // 0/1: which K-pair half of the wave
    const long row0  = (long)blockIdx.x * ROWS_PB;

    // 8 accumulator tiles: 4 expert tiles x {g, s_raw}; 16x16 f32 each (8 VGPRs)
    v8f accg[4], accs[4];
    #pragma unroll
    for (int t = 0; t < 4; ++t)
        #pragma unroll
        for (int j = 0; j < 8; ++j) { accg[t][j] = 0.0f; accs[t][j] = 0.0f; }

    // ---------------- software-pipelined GEMM over K ----------------
    stage_chunk(x, Wg, Ws, row0, 0, sm.g.X[0], sm.g.Wg[0], sm.g.Ws[0], tid);

    for (int c = 0; c < NCHUNK; ++c) {
        const int cur = c & 1;
        if (c + 1 < NCHUNK) {
            // issue next chunk's DMA into the other buffer, then wait only for
            // the current chunk (<= OPS_PT outstanding = just-issued ops)
            stage_chunk(x, Wg, Ws, row0, (c + 1) * KC,
                        sm.g.X[cur ^ 1], sm.g.Wg[cur ^ 1], sm.g.Ws[cur ^ 1], tid);
            // keep L2 warm two chunks ahead (global_prefetch_b8)
            if (c + 2 < NCHUNK)
                __builtin_prefetch(&x[(row0 + (tid & 127)) * (long)ND +
                                      (c + 2) * KC + (tid >> 7) * 16], 0, 0);
#if USE_ASYNC_LDS
            __builtin_amdgcn_s_wait_asynccnt(OPS_PT);
#endif
        } else {
#if USE_ASYNC_LDS
            __builtin_amdgcn_s_wait_asynccnt(0);
#endif
        }
        __syncthreads();   // current buffer fully staged for all waves

        const float* sX   = sm.g.X[cur];
        const float* sWgc = sm.g.Wg[cur];
        const float* sWsc = sm.g.Ws[cur];

        // 8 K-steps of 4; per step: 1 A-frag, 8 B-frags, 8 f32 WMMAs
        #pragma unroll
        for (int kk = 0; kk < KC; kk += 4) {
            // A 16x4 f32: lane(M=m16, half khalf) holds K = 2*khalf + {0,1}
            const v2f a = *(const v2f*)&sX[(wave * 16 + m16) * XST + kk + 2 * khalf];
            #pragma unroll
            for (int t = 0; t < 4; ++t) {
                // B 4x16 f32: lane(N=m16) holds K = 2*khalf + {0,1}; B(k,n)=W[e0+n][k]
                const v2f bg = *(const v2f*)&sWgc[(t * 16 + m16) * WST + kk + 2 * khalf];
                const v2f bs = *(const v2f*)&sWsc[(t * 16 + m16) * WST + kk + 2 * khalf];
                accg[t] = __builtin_amdgcn_wmma_f32_16x16x4_f32(
                    false, a, false, bg, (short)0, accg[t], false, false);
                accs[t] = __builtin_amdgcn_wmma_f32_16x16x4_f32(
                    false, a, false, bs, (short)0, accs[t], false, false);
            }
        }
        __syncthreads();   // all waves done reading before buffer is re-staged
    }

    // ---------------- dump fragments -> row-major LDS ----------------
    // (GEMM staging is dead; epilogue arrays alias it via the union)
    #pragma unroll
    for (int t = 0; t < 4; ++t)
        #pragma unroll
        for (int r = 0; r < 8; ++r) {
            const int row = wave * 16 + r + 8 * khalf;  // C/D: M = r + 8*half
            const int col = t * 16 + m16;               // N = lane & 15
            sm.e.H[row * EST + col] = accg[t][r];       // g (temporarily)
            sm.e.P[row * EST + col] = accs[t][r];       // s_raw (temporarily)
        }
    __syncthreads();

    // ---------------- per-row epilogue (thread t owns row t) ----------------
    float r_mmax = 0.0f, r_sum = 1.0f, r_eps = 0.0f;
    int   r_i0 = -1, r_i1 = -1;
    if (tid < ROWS_PB) {
        const int r = tid;
        r_eps = eps[row0 + r];
        // pass 1: scale = softplus(s_raw); h = g + eps*scale; track top-3
        float t0 = NEG_INF, t1 = NEG_INF, t2 = NEG_INF;
        int   i0 = 0, i1 = 0;
        for (int e = 0; e < NE; ++e) {
            const float g  = sm.e.H[r * EST + e];
            const float sr = sm.e.P[r * EST + e];
            const float sc = fmaxf(sr, 0.0f) + log1pf(expf(-fabsf(sr)));  // stable softplus
            const float h  = fmaf(r_eps, sc, g);
            sm.e.H[r * EST + e] = h;
            sm.e.P[r * EST + e] = sc;
            if (h > t0)      { t2 = t1; t1 = t0; i1 = i0; t0 = h; i0 = e; }
            else if (h > t1) { t2 = t1; t1 = h; i1 = e; }
            else if (h > t2) { t2 = h; }
        }
        // pass 2: thr = (h > t2) ? t2 : t1; P = Phi((g - thr)/scale); softmax max
        float mmax = NEG_INF;
        for (int e = 0; e < NE; ++e) {
            const float h   = sm.e.H[r * EST + e];
            const float sc  = sm.e.P[r * EST + e];
            const float g   = h - r_eps * sc;
            const float thr = (h > t2) ? t2 : t1;
            const float z   = (g - thr) / sc;
            sm.e.P[r * EST + e] = 0.5f * (1.0f + erff(z * 0.70710678118654752f));
            const float masked  = (e == i0 || e == i1) ? h : BETA_F;
            mmax = fmaxf(mmax, masked);
        }
        // pass 3: softmax denominator
        float se = 0.0f;
        for (int e = 0; e < NE; ++e) {
            const float h      = sm.e.H[r * EST + e];
            const float masked = (e == i0 || e == i1) ? h : BETA_F;
            se += expf(masked - mmax);
        }
        r_mmax = mmax; r_sum = se; r_i0 = i0; r_i1 = i1;
    }
    __syncthreads();

    // ---------------- per-block column sums -> global atomics ----------------
    if (tid < NE) {
        float hs = 0.0f, ps = 0.0f;
        for (int r = 0; r < ROWS_PB; ++r) {
            hs += sm.e.H[r * EST + tid];
            ps += sm.e.P[r * EST + tid];
        }
        atomicAdd(&wsum[tid], hs);
        atomicAdd(&wsum[NE + tid], ps);
    }
    __syncthreads();

    // ---------------- gates into LDS, then coalesced store ----------------
    if (tid < ROWS_PB) {
        const int r = tid;
        const float inv = 1.0f / r_sum;
        for (int e = 0; e < NE; ++e) {
            const float h      = sm.e.H[r * EST + e];
            const float masked = (e == r_i0 || e == r_i1) ? h : BETA_F;
            sm.e.P[r * EST + e] = expf(masked - r_mmax) * inv;
        }
    }
    __syncthreads();
    for (int i = tid; i < ROWS_PB * NE; i += 256) {
        const int r = i >> 6, e = i & 63;
        gates[row0 * NE + i] = sm.e.P[r * EST + e];
    }
}

__global__ void router_finalize(const float* __restrict__ wsum,
                                float* __restrict__ aux_out) {
    if (threadIdx.x == 0) {
        float mh = 0.0f, mp = 0.0f;
        for (int e = 0; e < NE; ++e) { mh += wsum[e]; mp += wsum[NE + e]; }
        mh *= (1.0f / NE); mp *= (1.0f / NE);
        float vh = 0.0f, vp = 0.0f;
        for (int e = 0; e < NE; ++e) {
            const float dh = wsum[e] - mh;       vh += dh * dh;
            const float dp = wsum[NE + e] - mp;  vp += dp * dp;
        }
        vh *= (1.0f / (NE - 1));   // ddof = 1
        vp *= (1.0f / (NE - 1));
        const float l_imp  = vh / (mh * mh);
        const float l_load = vp / (mp * mp);
        aux_out[0] = 0.1f * l_imp + 0.1f * l_load;
    }
}

extern "C" void kernel_launch(void* const* d_in, const int* in_sizes, int n_in,
                              void* d_out, int out_size, void* d_ws, size_t ws_size,
                              hipStream_t stream) {
    const float* x   = (const float*)d_in[0];
    const float* Wg  = (const float*)d_in[1];
    const float* Ws  = (const float*)d_in[2];
    const float* eps = (const float*)d_in[3];
    float* gates = (float*)d_out;                          // B*E floats
    float* aux   = gates + (size_t)NB * NE;                // final scalar
    float* wsum  = (float*)d_ws;                           // 128 f32 accumulators

    router_zero_ws<<<1, 128, 0, stream>>>(wsum);           // must re-zero every call
    router_main<<<NB / ROWS_PB, 256, 0, stream>>>(x, Wg, Ws, eps, gates, wsum);
    router_finalize<<<1, 32, 0, stream>>>(wsum, aux);
}